// Decoder_90374701842993
// MI455X (gfx1250) — compile-verified
//
#include <hip/hip_runtime.h>

// ---------------------------------------------------------------------------
// Factorized softmax-resample compositor for gfx1250 (MI455X).
//
// Math: A[p,k] = er[r,i]*ec[c,j]/(Sr[r]*Sc[c])  (separable softmax), so
//   R[ch,r,c] = ( (er @ img[ch]) @ ec^T )[r,c] / (Sr[r]*Sc[c])
// Two small f32 GEMMs per (batch,channel) using V_WMMA_F32_16X16X4_F32.
// ---------------------------------------------------------------------------

typedef __attribute__((ext_vector_type(2))) float v2f;
typedef __attribute__((ext_vector_type(8))) float v8f;

#define HH   256
#define WW   256
#define SS   64
#define NIMG 14

struct BoxParams { float x1, x2, y1, y2; int valid; int idx; int pad0, pad1; };

// ---------------------------------------------------------------------------
// Kernel 1: per-batch params (argmax image index, rounded box, validity).
// ---------------------------------------------------------------------------
__global__ void k_params(const float* __restrict__ X, BoxParams* __restrict__ P) {
    int b = threadIdx.x;
    if (b < 2) {
        const float* d = X + b * 19;   // X is (2,1,19); Xf = X[:, :1]
        int best = 0; float bv = d[5];
        for (int i = 1; i < NIMG; ++i) {       // first-max wins, like jnp.argmax
            float v = d[5 + i];
            if (v > bv) { bv = v; best = i; }
        }
        float x1 = rintf(d[0] * 256.0f);       // jnp.round = round-half-even
        float x2 = rintf(d[1] * 256.0f);
        float y1 = rintf(d[2] * 256.0f);
        float y2 = rintf(d[3] * 256.0f);
        int valid = (x1 >= 0.0f) && (y1 >= 0.0f) && (x2 <= 256.0f) &&
                    (y2 <= 256.0f) && (x2 > x1) && (y2 > y1);
        BoxParams p; p.x1 = x1; p.x2 = x2; p.y1 = y1; p.y2 = y2;
        p.valid = valid; p.idx = best; p.pad0 = 0; p.pad1 = 0;
        P[b] = p;
    }
}

// ---------------------------------------------------------------------------
// Kernel 2: separable softmax weights along one axis.
//   er[b][r][i] = exp(m(r) - (r - s_i)^2),  m(r) = min_i (r - s_i)^2
//   Sr[b][r]    = sum_i er[b][r][i]
// One wave32 per (pixel coordinate, axis, batch); each lane handles i, i+32.
// ---------------------------------------------------------------------------
__global__ void k_axis(const BoxParams* __restrict__ P,
                       float* __restrict__ er, float* __restrict__ ec,
                       float* __restrict__ Sr, float* __restrict__ Sc) {
    int r    = blockIdx.x;      // pixel coordinate along this axis (0..255)
    int axis = blockIdx.y;      // 0: rows (x1,x2), 1: cols (y1,y2)
    int b    = blockIdx.z;
    int t    = threadIdx.x;     // 0..31 (wave32)

    BoxParams p = P[b];
    float a1  = axis ? p.y1 : p.x1;
    float a2  = axis ? p.y2 : p.x2;
    float box = a2 - a1;
    float fr  = (float)r;

    float s0 = box * (1.0f / 64.0f) * (float)t        + a1;
    float s1 = box * (1.0f / 64.0f) * (float)(t + 32) + a1;
    float d0 = (fr - s0) * (fr - s0);
    float d1 = (fr - s1) * (fr - s1);

    float m = fminf(d0, d1);                      // wave-wide min (softmax max)
    for (int off = 16; off > 0; off >>= 1)
        m = fminf(m, __shfl_xor(m, off, 32));

    float e0 = expf(m - d0);                      // in (0, 1]
    float e1 = expf(m - d1);

    float s = e0 + e1;                            // wave-wide sum
    for (int off = 16; off > 0; off >>= 1)
        s += __shfl_xor(s, off, 32);

    float* E = (axis ? ec : er) + ((size_t)b * 256 + r) * 64;
    E[t]      = e0;
    E[t + 32] = e1;
    if (t == 0) (axis ? Sc : Sr)[b * 256 + r] = s;
}

// ---------------------------------------------------------------------------
// WMMA fragment helpers for V_WMMA_F32_16X16X4_F32 (wave32):
//   A 16x4 : lane L -> row M=L&15, VGPR v holds K = 2*(L>>4) + v
//   B 4x16 : lane L -> col N=L&15, VGPR v holds K = 2*(L>>4) + v
//   C/D    : lane L -> col N=L&15, VGPR v holds row M = v + 8*(L>>4)
// ---------------------------------------------------------------------------

// Kernel 3: T[b][ch][r][j] = sum_i er[b][r][i] * img[idx][ch][i][j]
// M=256 (r), N=64 (j), K=64 (i). One wave per 16x16 tile: grid (16,4,6).
__global__ void k_gemm1(const BoxParams* __restrict__ P,
                        const float* __restrict__ er,
                        const float* __restrict__ images,
                        float* __restrict__ T) {
    int mt = blockIdx.x;            // 0..15  row tile
    int nt = blockIdx.y;            // 0..3   col tile
    int z  = blockIdx.z;            // 0..5
    int b = z / 3, ch = z % 3;
    int idx = P[b].idx;

    const float* A = er + (size_t)b * 256 * 64;                     // [256][64]
    const float* B = images + ((size_t)idx * 4 + ch) * 64 * 64;     // [64][64]

    int lane = threadIdx.x;
    int half = lane >> 4;
    int l    = lane & 15;
    int mA   = mt * 16 + l;         // A row for this lane
    int nB   = nt * 16 + l;         // B col for this lane

    v8f acc = {};
    #pragma unroll
    for (int k0 = 0; k0 < 64; k0 += 4) {
        int ka = k0 + 2 * half;
        v2f a, bf;
        a.x  = A[mA * 64 + ka];
        a.y  = A[mA * 64 + ka + 1];
        bf.x = B[ka * 64 + nB];
        bf.y = B[(ka + 1) * 64 + nB];
        acc = __builtin_amdgcn_wmma_f32_16x16x4_f32(
            false, a, false, bf, (short)0, acc, false, false);
    }

    float* Tb = T + ((size_t)(b * 3 + ch)) * 256 * 64;
    int col = nt * 16 + l;
    #pragma unroll
    for (int v = 0; v < 8; ++v) {
        int row = mt * 16 + v + half * 8;
        Tb[row * 64 + col] = acc[v];
    }
}

// Kernel 4: U[r][c] = sum_j T[r][j] * ec[c][j], then epilogue
//   out = valid ? (inside ? U/(Sr[r]*Sc[c]) : 1.0) : 0.0
// M=256 (r), N=256 (c), K=64 (j). One wave per 16x16 tile: grid (16,16,6).
__global__ void k_gemm2(const BoxParams* __restrict__ P,
                        const float* __restrict__ T,
                        const float* __restrict__ ec,
                        const float* __restrict__ Sr,
                        const float* __restrict__ Sc,
                        float* __restrict__ out) {
    int mt = blockIdx.x;            // row tile (r)
    int nt = blockIdx.y;            // col tile (c)
    int z  = blockIdx.z;
    int b = z / 3, ch = z % 3;
    BoxParams p = P[b];

    const float* A = T  + ((size_t)(b * 3 + ch)) * 256 * 64;   // [256][64]
    const float* E = ec + (size_t)b * 256 * 64;                // [256][64]; B[k][n]=E[n*64+k]

    int lane = threadIdx.x;
    int half = lane >> 4;
    int l    = lane & 15;
    int mA   = mt * 16 + l;
    int nB   = nt * 16 + l;

    v8f acc = {};
    #pragma unroll
    for (int k0 = 0; k0 < 64; k0 += 4) {
        int ka = k0 + 2 * half;
        v2f a, bf;
        a.x  = A[mA * 64 + ka];
        a.y  = A[mA * 64 + ka + 1];
        bf.x = E[nB * 64 + ka];        // B is ec^T: column n = ec row n
        bf.y = E[nB * 64 + ka + 1];
        acc = __builtin_amdgcn_wmma_f32_16x16x4_f32(
            false, a, false, bf, (short)0, acc, false, false);
    }

    int   c   = nt * 16 + l;
    float fc  = (float)c;
    float scv = Sc[b * 256 + c];
    bool  cin = (fc >= p.y1) && (fc < p.y2);
    float* O  = out + ((size_t)(b * 3 + ch)) * 256 * 256;

    #pragma unroll
    for (int v = 0; v < 8; ++v) {
        int   r   = mt * 16 + v + half * 8;
        float fr  = (float)r;
        float val = acc[v] / (Sr[b * 256 + r] * scv);
        bool inside = cin && (fr >= p.x1) && (fr < p.x2);
        O[r * 256 + c] = p.valid ? (inside ? val : 1.0f) : 0.0f;
    }
}

// ---------------------------------------------------------------------------
// Workspace layout (bytes):
//   er : [2][256][64] f32  @ 0        (131072)
//   ec : [2][256][64] f32  @ 131072   (131072)
//   Sr : [2][256]     f32  @ 262144   (2048)
//   Sc : [2][256]     f32  @ 264192   (2048)
//   T  : [2][3][256][64]   @ 266240   (393216)
//   P  : BoxParams[2]      @ 659456   (64)
// Total ~660 KB.
// ---------------------------------------------------------------------------
extern "C" void kernel_launch(void* const* d_in, const int* in_sizes, int n_in,
                              void* d_out, int out_size, void* d_ws, size_t ws_size,
                              hipStream_t stream) {
    const float* X      = (const float*)d_in[0];   // (2,1,19) f32
    const float* images = (const float*)d_in[1];   // (14,4,64,64) f32
    float* out          = (float*)d_out;           // (2,3,256,256) f32

    char* ws = (char*)d_ws;
    float*     er = (float*)(ws + 0);
    float*     ec = (float*)(ws + 131072);
    float*     Sr = (float*)(ws + 262144);
    float*     Sc = (float*)(ws + 264192);
    float*     T  = (float*)(ws + 266240);
    BoxParams* P  = (BoxParams*)(ws + 659456);

    k_params<<<1, 32, 0, stream>>>(X, P);
    k_axis  <<<dim3(256, 2, 2), 32, 0, stream>>>(P, er, ec, Sr, Sc);
    k_gemm1 <<<dim3(16, 4, 6),  32, 0, stream>>>(P, er, images, T);
    k_gemm2 <<<dim3(16, 16, 6), 32, 0, stream>>>(P, T, ec, Sr, Sc, out);
}